// MainModule_3_30253749633603
// MI455X (gfx1250) — compile-verified
//
#include <hip/hip_runtime.h>
#include <stdint.h>

// ---------------------------------------------------------------------------
// Pyramid blend + connectivity stencil for MI455X (gfx1250, wave32).
// Latency/bandwidth-bound f32 stencil: no GEMM structure -> no WMMA.
// CDNA5-specific path used: async global->LDS tile staging (ASYNCcnt).
// ---------------------------------------------------------------------------

#define TILE 16
#define HALO 18            // TILE + 2 (one-pixel halo each side)
#define ROWF (HALO * 7)    // 126 valid floats per tile row
#define ROWP 128           // padded LDS row stride in floats

// gfx1250 async global->LDS copy, 4B per lane. VDST operand = LDS byte
// offset in a VGPR, VADDR = 64-bit global address in a VGPR pair.
__device__ __forceinline__ void async_g2l_b32(uint32_t lds_off, const void* gaddr) {
  asm volatile("global_load_async_to_lds_b32 %0, %1, off"
               :: "v"(lds_off), "v"(gaddr)
               : "memory");
}
__device__ __forceinline__ void wait_async0() {
  asm volatile("s_wait_asynccnt 0" ::: "memory");
}

// relu2(v) = 1 + elu(v) = (v > 0) ? 1+v : exp(v)
__device__ __forceinline__ float relu2f(float v) {
  return v > 0.0f ? 1.0f + v : __expf(v);
}

// ---------------------------------------------------------------------------
// Merge kernel: plot (H,W,7) with sqrt'ed weight, optionally blended with the
// nearest-upsampled previous part (H/2,W/2,7) whose weight is scaled by pp.
// ---------------------------------------------------------------------------
__global__ __launch_bounds__(256) void merge_kernel(
    const float* __restrict__ plot, const float* __restrict__ prev,
    float* __restrict__ out, int H, int W, const float* __restrict__ pp_p) {
  int idx = blockIdx.x * blockDim.x + threadIdx.x;
  int npix = H * W;
  if (idx >= npix) return;
  int y = idx / W;
  int x = idx - y * W;
  const float* b = plot + (size_t)idx * 7;
  float o[7];
  float wb = sqrtf(b[6]);
  if (prev == nullptr) {
#pragma unroll
    for (int c = 0; c < 6; ++c) o[c] = b[c];
    o[6] = wb;
  } else {
    const float* a = prev + ((size_t)(y >> 1) * (size_t)(W >> 1) + (x >> 1)) * 7;
    float wa = a[6] * pp_p[0];
    float ws = wa + wb;
    float inv = 1.0f / ws;
#pragma unroll
    for (int c = 0; c < 6; ++c) o[c] = (wa * a[c] + wb * b[c]) * inv;
    o[6] = log1pf(ws);
  }
  float* op = out + (size_t)idx * 7;
#pragma unroll
  for (int c = 0; c < 7; ++c) op[c] = o[c];
}

// ---------------------------------------------------------------------------
// Connectivity kernel: 16x16 output tile, 18x18x7 halo tile staged in LDS via
// async global->LDS loads. OOB taps are 1.0 (LDS pre-filled with 1.0).
// tap index k = (dy+1)*3 + (dx+1); reference xy[I][J] => dx=I-1, dy=J-1.
// ---------------------------------------------------------------------------

// TERM: channel R-1 of a tap times a selected weight; R==7 (ch 6) is the
// reference's empty slice -> contributes zero.
#define TERM(tp, CH, WS) ((CH) < 6 ? (tp)[CH] * wt[WS] : 0.0f)

#define DO_PAIR(T1, C1A, W1A, C1B, W1B, T2, C2A, W2A, C2B, W2B)               \
  do {                                                                        \
    const float* t1 = t[T1];                                                  \
    const float* t2 = t[T2];                                                  \
    float w1 = t1[6], w2 = t2[6];                                             \
    float s1 = TERM(t1, C1A, W1A) + TERM(t1, C1B, W1B);                       \
    float s2 = TERM(t2, C2A, W2A) + TERM(t2, C2B, W2B);                       \
    float g = relu2f(s1) * relu2f(s2);                                        \
    float w12 = w1 + w2;                                                      \
    wsum += g * w12;                                                          \
    _Pragma("unroll")                                                         \
    for (int c = 0; c < 6; ++c) csum[c] += g * (t1[c] * w1 + t2[c] * w2);     \
    csum[6] += g * w12 * w12;                                                 \
  } while (0)

template <bool FINAL>
__global__ __launch_bounds__(256) void conn_kernel(
    const float* __restrict__ in, float* __restrict__ out,
    int H, int W, const float* __restrict__ kp_p, const float* __restrict__ dp_p) {
  __shared__ float tile[HALO * ROWP];
  const int tid = threadIdx.x;
  const int x0 = blockIdx.x * TILE;
  const int y0 = blockIdx.y * TILE;

  // 1) boundary value 1.0 everywhere (interior overwritten by async loads)
  for (int i = tid; i < HALO * ROWP; i += 256) tile[i] = 1.0f;
  __syncthreads();

  // 2) async-copy the in-bounds part of the 18x18x7 halo tile into LDS
  for (int i = tid; i < HALO * ROWF; i += 256) {
    int r = i / ROWF;
    int c = i - r * ROWF;
    int gy = y0 - 1 + r;
    int gx = (x0 - 1) * 7 + c;  // float index within source row
    if (gy >= 0 && gy < H && gx >= 0 && gx < W * 7) {
      const float* gp = in + (size_t)gy * (size_t)(W * 7) + gx;
      uint32_t loff = (uint32_t)(uintptr_t)&tile[r * ROWP + c];
      async_g2l_b32(loff, gp);
    }
  }
  wait_async0();     // wave's own async loads landed in LDS
  __syncthreads();   // all waves' loads visible

  // 3) pull the 3x3 neighborhood (9 taps x 7 ch) into registers
  const int lx = tid & 15, ly = tid >> 4;
  float t[9][7];
#pragma unroll
  for (int jj = 0; jj < 3; ++jj)
#pragma unroll
    for (int ii = 0; ii < 3; ++ii)
#pragma unroll
      for (int c = 0; c < 7; ++c)
        t[jj * 3 + ii][c] = tile[(ly + jj) * ROWP + (lx + ii) * 7 + c];

  // weight table: 0=s2(0.5), 1=1.0, 2=d1, 3=d2, 4=s4(dp)
  const float kp = kp_p[0], dp = dp_p[0];
  const float SQ3H = 0.86602540378443864676f;
  float wt[5];
  wt[0] = 0.5f;
  wt[1] = 1.0f;
  wt[2] = 0.5f / (SQ3H + 0.5f) * kp;
  wt[3] = SQ3H / (SQ3H + 0.5f) * kp;
  wt[4] = dp;

  float csum[7];
#pragma unroll
  for (int c = 0; c < 7; ++c) csum[c] = t[4][c];  // center tap
  float wsum = 1.0f;

  // 16 pairs from the reference (tap, chA, wselA, chB, wselB) x 2
  DO_PAIR(0, 3, 0, 5, 0,   8, 4, 0, 6, 0);
  DO_PAIR(6, 3, 0, 6, 0,   2, 4, 0, 5, 0);
  DO_PAIR(1, 3, 1, 6, 0,   7, 4, 1, 6, 0);
  DO_PAIR(3, 5, 1, 6, 0,   5, 6, 1, 6, 0);
  DO_PAIR(1, 4, 3, 5, 2,   6, 3, 2, 6, 2);
  DO_PAIR(1, 4, 3, 6, 2,   8, 3, 2, 5, 2);
  DO_PAIR(0, 4, 3, 6, 2,   7, 3, 3, 5, 2);
  DO_PAIR(2, 4, 3, 5, 2,   7, 3, 3, 6, 2);
  DO_PAIR(2, 4, 2, 5, 3,   3, 3, 2, 6, 3);
  DO_PAIR(3, 4, 2, 6, 3,   8, 3, 2, 5, 3);
  DO_PAIR(0, 4, 2, 6, 3,   5, 3, 2, 5, 3);
  DO_PAIR(5, 4, 2, 5, 3,   6, 3, 2, 6, 3);
  DO_PAIR(1, 3, 4, 5, 4,   5, 4, 4, 6, 4);
  DO_PAIR(1, 3, 4, 6, 4,   1, 4, 4, 5, 4);
  DO_PAIR(3, 3, 4, 5, 4,   7, 4, 4, 6, 4);
  DO_PAIR(7, 3, 4, 6, 4,   5, 4, 4, 5, 4);

  float inv = 1.0f / wsum;
  int gx = x0 + lx, gy = y0 + ly;
  if (FINAL) {
    float* op = out + ((size_t)gy * W + gx) * 3;
#pragma unroll
    for (int c = 0; c < 3; ++c) op[c] = csum[c] * inv;
  } else {
    float* op = out + ((size_t)gy * W + gx) * 7;
#pragma unroll
    for (int c = 0; c < 7; ++c) op[c] = csum[c] * inv;
  }
}

// ---------------------------------------------------------------------------
// Host-side launch: 4 levels, smallest first, ping-pong in d_ws.
// ---------------------------------------------------------------------------
extern "C" void kernel_launch(void* const* d_in, const int* in_sizes, int n_in,
                              void* d_out, int out_size, void* d_ws, size_t ws_size,
                              hipStream_t stream) {
  (void)in_sizes; (void)n_in; (void)out_size; (void)ws_size;
  const float* plot0 = (const float*)d_in[0];  // 1024^2 x 7
  const float* plot1 = (const float*)d_in[1];  // 512^2  x 7
  const float* plot2 = (const float*)d_in[2];  // 256^2  x 7
  const float* plot3 = (const float*)d_in[3];  // 128^2  x 7
  const float* kp = (const float*)d_in[4];
  const float* pp = (const float*)d_in[5];
  const float* dp = (const float*)d_in[6];
  float* out  = (float*)d_out;
  float* bufA = (float*)d_ws;                       // merged plane (<= 28 MiB)
  float* bufB = bufA + (size_t)1024 * 1024 * 7;     // part plane   (<= 28 MiB)

  struct Level { const float* plot; int N; };
  const Level levels[4] = {{plot3, 128}, {plot2, 256}, {plot1, 512}, {plot0, 1024}};

  for (int l = 0; l < 4; ++l) {
    const int N = levels[l].N;
    const int npix = N * N;
    const float* prev = (l == 0) ? nullptr : bufB;
    merge_kernel<<<(npix + 255) / 256, 256, 0, stream>>>(
        levels[l].plot, prev, bufA, N, N, pp);
    dim3 grid(N / TILE, N / TILE);
    if (l == 3)
      conn_kernel<true><<<grid, 256, 0, stream>>>(bufA, out, N, N, kp, dp);
    else
      conn_kernel<false><<<grid, 256, 0, stream>>>(bufA, bufB, N, N, kp, dp);
  }
}